// HubertTimeMask2D_14946486190270
// MI455X (gfx1250) — compile-verified
//
#include <hip/hip_runtime.h>
#include <stdint.h>

// Problem constants (from reference):
//   x: (B=8, C=64, T=2048, F=80) fp32;  mask_embedding: (1,64,1,1) fp32
//   num_mask = int(0.065*2048/10) = 13 spans of length 10 per (b) row.
#define B_DIM 8
#define C_DIM 64
#define T_DIM 2048
#define F_DIM 80
#define NUM_MASK 13
#define MASK_LEN 10
#define F4_PER_ROW (F_DIM / 4)                 // 20 float4 per (b,c,t) row
#define PLANE_F4 (T_DIM * F4_PER_ROW)          // 40960 float4 per (b,c) plane
#define X_ELEMS ((size_t)B_DIM * C_DIM * T_DIM * F_DIM)   // 83,886,080
#define MASK_ELEMS (B_DIM * T_DIM)             // 16,384

typedef __attribute__((ext_vector_type(4))) float v4f;
typedef __attribute__((ext_vector_type(4))) unsigned int v4u;

// ---------------- Kernel 1 (hot, first in file for disasm): NT streaming select
// 640 MB of traffic at 23.3 TB/s ~= 28 us floor; zero FLOPs -> pure BW kernel.
// grid = (PLANE_F4/256, B*C); block = 256 (8 wave32). One v4f per lane.
// Branchless: issue the b128 NT load, the mask-byte load and the scalar
// embedding load back-to-back, then 4x v_cndmask, then b128 NT store.
// NT on the streamed 320MB in + 320MB out (touched once, working set > L2);
// mask byte (reused 5120x) and embedding (uniform) stay regular-temporal.
__global__ void __launch_bounds__(256)
apply_kernel(const float* __restrict__ x,
             const float* __restrict__ emb,
             const unsigned char* __restrict__ mask,
             float* __restrict__ out) {
    int plane = blockIdx.y;            // b*C + c  (uniform per block)
    int b = plane >> 6;
    int c = plane & (C_DIM - 1);
    int tid = blockIdx.x * 256 + threadIdx.x;   // 0 .. PLANE_F4-1
    int t = tid / F4_PER_ROW;

    size_t off = (size_t)plane * (T_DIM * (size_t)F_DIM) + (size_t)tid * 4;
    v4f v = __builtin_nontemporal_load((const v4f*)(x + off));
    unsigned m = mask[b * T_DIM + t];  // cached byte
    float e = emb[c];                  // uniform -> s_load
    v.x = m ? e : v.x;                 // v_cndmask x4, no EXEC branching
    v.y = m ? e : v.y;
    v.z = m ? e : v.z;
    v.w = m ? e : v.w;
    __builtin_nontemporal_store(v, (v4f*)(out + off));
}

// ---------------- Threefry-2x32-20 (JAX PRNG core) ----------------
__device__ __forceinline__ void threefry2x32(uint32_t k0, uint32_t k1,
                                             uint32_t x0, uint32_t x1,
                                             uint32_t* o0, uint32_t* o1) {
    const uint32_t rot[8] = {13, 15, 26, 6, 17, 29, 16, 24};
    uint32_t ks[3] = {k0, k1, k0 ^ k1 ^ 0x1BD11BDAu};
    x0 += ks[0];
    x1 += ks[1];
#pragma unroll
    for (int r = 0; r < 5; ++r) {
#pragma unroll
        for (int i = 0; i < 4; ++i) {
            uint32_t R = rot[(r & 1) * 4 + i];
            x0 += x1;
            x1 = (x1 << R) | (x1 >> (32 - R));
            x1 ^= x0;
        }
        x0 += ks[(r + 1) % 3];
        x1 += ks[(r + 2) % 3] + (uint32_t)(r + 1);
    }
    *o0 = x0;
    *o1 = x1;
}

// ---------------- Kernel 2: zero mask bytes (b128) + stamp spans ----------
// One block, 256 threads. 16KB zeroed with 4 b128 stores/lane; 104 span
// generators. Overlapping spans all write 1 -> benign races, deterministic.
__global__ void __launch_bounds__(256)
genmask_kernel(unsigned char* __restrict__ mask) {
    int tid = threadIdx.x;
    v4u z = {0u, 0u, 0u, 0u};
#pragma unroll
    for (int i = 0; i < MASK_ELEMS / (256 * 16); ++i)   // 4 iters
        *(v4u*)(mask + (i * 256 + tid) * 16) = z;
    __syncthreads();
    if (tid < B_DIM * NUM_MASK) {
        int b = tid / NUM_MASK;
        int s = tid % NUM_MASK;
        uint32_t r0, r1;
        // key(42) folded per (b, span); counter = flat span index.
        threefry2x32(0u, 42u, (uint32_t)b, (uint32_t)(b * NUM_MASK + s), &r0, &r1);
        int start = (int)(r0 % (uint32_t)(T_DIM - MASK_LEN + 1));
        unsigned char* row = mask + b * T_DIM + start;
#pragma unroll
        for (int j = 0; j < MASK_LEN; ++j) row[j] = 1;
    }
}

// ---------------- Kernel 3: mask -> float via CDNA5 async-to-LDS ----------
// 32 blocks x 32 lanes (1 wave). Each lane async-copies 16 mask bytes
// (global_load_async_to_lds_b128), waits on ASYNCcnt, reads them back from
// LDS and emits 16 floats. Exercises the gfx1250 async path + ds + wait.
__global__ void __launch_bounds__(32)
maskout_kernel(const unsigned char* __restrict__ mask, float* __restrict__ mout) {
    __shared__ __align__(16) unsigned char smem[512];
    unsigned lane = threadIdx.x;

    // 32-bit LDS byte offset of this lane's 16-byte slot.
    unsigned lds_off =
        (unsigned)(size_t)((__attribute__((address_space(3))) unsigned char*)smem) +
        lane * 16u;
    unsigned long long gaddr =
        (unsigned long long)(mask + (size_t)blockIdx.x * 512u + lane * 16u);

    asm volatile("global_load_async_to_lds_b128 %0, %1, off"
                 :: "v"(lds_off), "v"(gaddr)
                 : "memory");
    asm volatile("s_wait_asynccnt 0" ::: "memory");
    __syncthreads();

    size_t base = (size_t)blockIdx.x * 512u + lane * 16u;
#pragma unroll
    for (int q = 0; q < 4; ++q) {
        v4f f;
        f.x = smem[lane * 16 + q * 4 + 0] ? 1.0f : 0.0f;
        f.y = smem[lane * 16 + q * 4 + 1] ? 1.0f : 0.0f;
        f.z = smem[lane * 16 + q * 4 + 2] ? 1.0f : 0.0f;
        f.w = smem[lane * 16 + q * 4 + 3] ? 1.0f : 0.0f;
        *(v4f*)(mout + base + q * 4) = f;
    }
}

extern "C" void kernel_launch(void* const* d_in, const int* in_sizes, int n_in,
                              void* d_out, int out_size, void* d_ws, size_t ws_size,
                              hipStream_t stream) {
    const float* x   = (const float*)d_in[0];        // (8,64,2048,80) fp32
    const float* emb = (const float*)d_in[1];        // (1,64,1,1) fp32
    float* out       = (float*)d_out;                // [x_masked | mask-as-float]
    unsigned char* mask_ws = (unsigned char*)d_ws;   // 16KB scratch

    (void)in_sizes; (void)n_in; (void)out_size; (void)ws_size;

    // 1) Build the (B,T) span mask in workspace (deterministic every call).
    genmask_kernel<<<dim3(1), dim3(256), 0, stream>>>(mask_ws);

    // 2) Stream the 640MB select: grid.y = (b,c) plane, grid.x tiles T*F/4.
    apply_kernel<<<dim3(PLANE_F4 / 256, B_DIM * C_DIM), dim3(256), 0, stream>>>(
        x, emb, mask_ws, out);

    // 3) Materialize the mask output (second tuple element) after x_masked.
    maskout_kernel<<<dim3(MASK_ELEMS / 512), dim3(32), 0, stream>>>(
        mask_ws, out + X_ELEMS);
}